// RecGCN_12000138625507
// MI455X (gfx1250) — compile-verified
//
#include <hip/hip_runtime.h>
#include <math.h>

typedef __attribute__((ext_vector_type(2))) float v2f;
typedef __attribute__((ext_vector_type(8))) float v8f;

#define F_IN 100
#define F_OUT 8

// ---------------------------------------------------------------- utilities
__global__ void fill_f32(float* p, float v, int n) {
    int i = blockIdx.x * blockDim.x + threadIdx.x;
    if (i < n) p[i] = v;
}

__global__ void rsqrt_inplace(float* p, int n) {
    int i = blockIdx.x * blockDim.x + threadIdx.x;
    if (i < n) p[i] = rsqrtf(p[i]);   // deg >= 1 always (self loop), no guard needed
}

// ----------------------------------------------------- degree scatter (edge-parallel)
__global__ void degree_kernel(const long long* __restrict__ adj, int E,
                              float* __restrict__ deg) {
    int e = blockIdx.x * blockDim.x + threadIdx.x;
    if (e >= E) return;
    long long d = adj[(size_t)E + e];          // dst row of edge_index
    unsafeAtomicAdd(deg + d, 1.0f);
}

// ----------------------------------------------------- h = x @ W via V_WMMA_F32_16X16X4_F32
// One wave per 16-row tile. W padded to 16 cols (cols 8..15 = 0) in LDS.
__global__ __launch_bounds__(128) void gcn_gemm_wmma(const float* __restrict__ x,
                                                     const float* __restrict__ W,
                                                     float* __restrict__ h,
                                                     int n_nodes) {
    __shared__ float Ws[F_IN * 16];
    const int tid = threadIdx.x;
    for (int i = tid; i < F_IN * 16; i += 128) {
        int k = i >> 4, n = i & 15;
        Ws[i] = (n < F_OUT) ? W[k * F_OUT + n] : 0.0f;
    }
    __syncthreads();

    const int wave = tid >> 5;
    const int lane = tid & 31;
    const int row0 = (blockIdx.x * 4 + wave) * 16;
    if (row0 >= n_nodes) return;               // wave-uniform: EXEC stays all-ones

    const int m  = lane & 15;                  // M (A) / N (B) position
    const int kh = (lane >> 4) * 2;            // lane-half K offset: 0 or 2

    const float* xrow = x + (size_t)(row0 + m) * F_IN;

    v8f c = {};
    #pragma unroll 5
    for (int k0 = 0; k0 < F_IN; k0 += 4) {
        v2f a, b;
        // A 16x4 f32: VGPR0 = K=kh, VGPR1 = K=kh+1 (lanes 0-15: K 0/1, lanes 16-31: K 2/3)
        a.x = xrow[k0 + kh];
        a.y = xrow[k0 + kh + 1];
        // B 4x16 f32: VGPR0 lanes0-15 = row K=k0, lanes16-31 = row K=k0+2; VGPR1 = +1
        b.x = Ws[(k0 + kh)     * 16 + m];
        b.y = Ws[(k0 + kh + 1) * 16 + m];
        c = __builtin_amdgcn_wmma_f32_16x16x4_f32(false, a, false, b,
                                                  (short)0, c, false, false);
    }

    // C/D layout: VGPR r -> M = r (lanes 0-15) or r+8 (lanes 16-31), N = lane&15
    if (m < F_OUT) {
        const int rbase = row0 + (lane >> 4) * 8;
        #pragma unroll
        for (int r = 0; r < 8; ++r)
            h[(size_t)(rbase + r) * F_OUT + m] = c[r];
    }
}

// ----------------------------------------------------- self-loop term: agg = h * dinv^2
__global__ void self_loop_init(const float* __restrict__ h,
                               const float* __restrict__ dinv,
                               float* __restrict__ agg, int nf) {
    int i = blockIdx.x * blockDim.x + threadIdx.x;
    if (i >= nf) return;
    float di = dinv[i >> 3];
    agg[i] = h[i] * di * di;
}

// ----------------------------------------------------- edge messages (scatter-add, L2 atomics)
__global__ void message_kernel(const long long* __restrict__ adj, int E,
                               const float* __restrict__ h,
                               const float* __restrict__ dinv,
                               float* __restrict__ agg) {
    int e = blockIdx.x * blockDim.x + threadIdx.x;
    if (e >= E) return;
    long long s = adj[e];
    long long d = adj[(size_t)E + e];
    float norm = dinv[s] * dinv[d];
    const float4* hs = (const float4*)(h + (size_t)s * F_OUT);
    float4 h0 = hs[0], h1 = hs[1];
    float* ag = agg + (size_t)d * F_OUT;
    unsafeAtomicAdd(ag + 0, h0.x * norm);
    unsafeAtomicAdd(ag + 1, h0.y * norm);
    unsafeAtomicAdd(ag + 2, h0.z * norm);
    unsafeAtomicAdd(ag + 3, h0.w * norm);
    unsafeAtomicAdd(ag + 4, h1.x * norm);
    unsafeAtomicAdd(ag + 5, h1.y * norm);
    unsafeAtomicAdd(ag + 6, h1.z * norm);
    unsafeAtomicAdd(ag + 7, h1.w * norm);
}

// ----------------------------------------------------- bias + tanh, in place
__global__ void finalize_kernel(float* __restrict__ agg,
                                const float* __restrict__ bias, int nf) {
    int i = blockIdx.x * blockDim.x + threadIdx.x;
    if (i >= nf) return;
    agg[i] = tanhf(agg[i] + bias[i & 7]);
}

// ----------------------------------------------------- decode: per-pair 8-dim dot
__global__ void score_kernel(const float* __restrict__ uemb,
                             const float* __restrict__ vemb,
                             const long long* __restrict__ uidx,
                             const long long* __restrict__ vidx,
                             float* __restrict__ out, int P) {
    int p = blockIdx.x * blockDim.x + threadIdx.x;
    if (p >= P) return;
    long long u = uidx[p], v = vidx[p];
    const float4* up = (const float4*)(uemb + (size_t)u * F_OUT);
    const float4* vp = (const float4*)(vemb + (size_t)v * F_OUT);
    float4 u0 = up[0], u1 = up[1];
    float4 v0 = vp[0], v1 = vp[1];
    out[p] = u0.x * v0.x + u0.y * v0.y + u0.z * v0.z + u0.w * v0.w +
             u1.x * v1.x + u1.y * v1.y + u1.z * v1.z + u1.w * v1.w;
}

// ================================================================ launcher
extern "C" void kernel_launch(void* const* d_in, const int* in_sizes, int n_in,
                              void* d_out, int out_size, void* d_ws, size_t ws_size,
                              hipStream_t stream) {
    const float*     x_user   = (const float*)d_in[0];
    const long long* adj_user = (const long long*)d_in[1];
    const float*     x_item   = (const float*)d_in[2];
    const long long* adj_item = (const long long*)d_in[3];
    const long long* user_idx = (const long long*)d_in[4];
    const long long* item_idx = (const long long*)d_in[5];
    const float*     W_user   = (const float*)d_in[6];
    const float*     b_user   = (const float*)d_in[7];
    const float*     W_item   = (const float*)d_in[8];
    const float*     b_item   = (const float*)d_in[9];
    float* score = (float*)d_out;

    const int n_nodes = in_sizes[0] / F_IN;     // 100000
    const int E       = in_sizes[1] / 2;        // 3.2M
    const int P       = in_sizes[4];            // 1M
    const int nf      = n_nodes * F_OUT;

    // workspace layout (floats): h_user, h_item, agg_user, agg_item, dinv_user, dinv_item
    float* ws = (float*)d_ws;
    float* h_user    = ws;
    float* h_item    = h_user + (size_t)nf;
    float* agg_user  = h_item + (size_t)nf;
    float* agg_item  = agg_user + (size_t)nf;
    float* dinv_user = agg_item + (size_t)nf;
    float* dinv_item = dinv_user + (size_t)n_nodes;

    const int T = 256;
    dim3 gN((n_nodes + T - 1) / T), gE((E + T - 1) / T),
         gF((nf + T - 1) / T),     gP((P + T - 1) / T);
    const int tiles = (n_nodes + 15) / 16;
    dim3 gG((tiles + 3) / 4);

    // 1. degrees (init to 1.0 for self-loop), then dinv = rsqrt(deg)
    fill_f32<<<gN, T, 0, stream>>>(dinv_user, 1.0f, n_nodes);
    fill_f32<<<gN, T, 0, stream>>>(dinv_item, 1.0f, n_nodes);
    degree_kernel<<<gE, T, 0, stream>>>(adj_user, E, dinv_user);
    degree_kernel<<<gE, T, 0, stream>>>(adj_item, E, dinv_item);
    rsqrt_inplace<<<gN, T, 0, stream>>>(dinv_user, n_nodes);
    rsqrt_inplace<<<gN, T, 0, stream>>>(dinv_item, n_nodes);

    // 2. h = x @ W via WMMA
    gcn_gemm_wmma<<<gG, 128, 0, stream>>>(x_user, W_user, h_user, n_nodes);
    gcn_gemm_wmma<<<gG, 128, 0, stream>>>(x_item, W_item, h_item, n_nodes);

    // 3. aggregation: self-loop term + edge scatter-add
    self_loop_init<<<gF, T, 0, stream>>>(h_user, dinv_user, agg_user, nf);
    self_loop_init<<<gF, T, 0, stream>>>(h_item, dinv_item, agg_item, nf);
    message_kernel<<<gE, T, 0, stream>>>(adj_user, E, h_user, dinv_user, agg_user);
    message_kernel<<<gE, T, 0, stream>>>(adj_item, E, h_item, dinv_item, agg_item);

    // 4. bias + tanh
    finalize_kernel<<<gF, T, 0, stream>>>(agg_user, b_user, nf);
    finalize_kernel<<<gF, T, 0, stream>>>(agg_item, b_item, nf);

    // 5. decode
    score_kernel<<<gP, T, 0, stream>>>(agg_user, agg_item, user_idx, item_idx, score, P);
}